// SparseMultiHeadAttention_47330539602118
// MI455X (gfx1250) — compile-verified
//
#include <hip/hip_runtime.h>
#include <hip/hip_bf16.h>

// ---------------------------------------------------------------------------
// Sparse multi-head graph attention for MI455X (gfx1250, wave32).
// Dense projections use V_WMMA_F32_16X16X4_F32 (fp32 WMMA, full precision).
// Segment softmax via atomic-max (monotone uint mapping) + atomic-add.
// ---------------------------------------------------------------------------

typedef __attribute__((ext_vector_type(2))) float v2f;
typedef __attribute__((ext_vector_type(8))) float v8f;

#define NNODES 100000
#define NEDGES 1600000
#define EMB    128
#define NHEADS 8
#define HDIM   16
#define EDIM   6

// ---------------------------------------------------------------------------
// C[M x 128] = A[M x 128] @ W[128 x 128] (+ bias), M multiple of 16.
// Block = 256 threads = 8 waves; each wave owns one 16-wide column tile.
// fp32 WMMA 16x16x4, K loop of 32 steps.
// A layout (ISA 7.12.2): lanes 0-15 rows M0-15 hold K=k0,k0+1; lanes 16-31
// hold K=k0+2,k0+3. B mirrors (rows striped across lanes within a VGPR).
// C/D: VGPR j -> row j (lanes 0-15) / row j+8 (lanes 16-31), col = lane&15.
// ---------------------------------------------------------------------------
__global__ __launch_bounds__(256) void gemm128_wmma(
    const float* __restrict__ A, const float* __restrict__ W,
    const float* __restrict__ bias, float* __restrict__ C) {
  __shared__ float As[16][132];  // pad 128->132: kills bank conflicts
  const int tid = threadIdx.x;
  const int r0  = blockIdx.x * 16;

  // cooperative load of the 16x128 A tile (512 float4 / 256 threads)
  const float4* Ag = (const float4*)(A + (size_t)r0 * EMB);
  for (int i = tid; i < 512; i += 256) {
    const int row = i >> 5;
    const int c4  = i & 31;
    float4 v = Ag[row * 32 + c4];
    As[row][c4 * 4 + 0] = v.x;
    As[row][c4 * 4 + 1] = v.y;
    As[row][c4 * 4 + 2] = v.z;
    As[row][c4 * 4 + 3] = v.w;
  }
  __syncthreads();

  const int wave  = tid >> 5;
  const int lane  = tid & 31;
  const int n0    = wave * 16;      // column tile for this wave
  const int rc    = lane & 15;      // A row / B col for this lane
  const int khalf = (lane >> 4) * 2;

  v8f acc = {};
#pragma unroll
  for (int kk = 0; kk < 32; ++kk) {
    const int k0 = kk * 4 + khalf;
    v2f a, b;
    a.x = As[rc][k0];
    a.y = As[rc][k0 + 1];
    b.x = W[(size_t)k0       * EMB + n0 + rc];
    b.y = W[(size_t)(k0 + 1) * EMB + n0 + rc];
    acc = __builtin_amdgcn_wmma_f32_16x16x4_f32(
        /*neg_a=*/false, a, /*neg_b=*/false, b,
        /*c_mod=*/(short)0, acc, /*reuse_a=*/false, /*reuse_b=*/false);
  }

  const int mbase = (lane >> 4) * 8;
  const float bv  = bias ? bias[n0 + rc] : 0.0f;
#pragma unroll
  for (int j = 0; j < 8; ++j)
    C[(size_t)(r0 + mbase + j) * EMB + n0 + rc] = acc[j] + bv;
}

// order-preserving float <-> uint mapping for atomicMax on floats
__device__ __forceinline__ unsigned fmap(float f) {
  unsigned u = __float_as_uint(f);
  return (u & 0x80000000u) ? ~u : (u | 0x80000000u);
}
__device__ __forceinline__ float funmap(unsigned u) {
  return (u & 0x80000000u) ? __uint_as_float(u ^ 0x80000000u)
                           : __uint_as_float(~u);
}

// ---------------------------------------------------------------------------
// Pass 1: per-edge scores. 8 lanes per edge (one per head); 4 edges/wave,
// 32 edges/block. Edge-MLP 6->128 (exact GELU) ->8 split 16 hidden units per
// lane, cross-lane reduced with shfl_xor. QK dot per head is a coalesced
// 64B read per lane (8 lanes cover the full 512B Q/K row).
// ---------------------------------------------------------------------------
__global__ __launch_bounds__(256) void edge_score_kernel(
    const long long* __restrict__ ei, const float* __restrict__ edge_attr,
    const float* __restrict__ Q, const float* __restrict__ K,
    const float* __restrict__ We1, const float* __restrict__ be1,
    const float* __restrict__ We2, const float* __restrict__ be2,
    float* __restrict__ attn, unsigned* __restrict__ segmax) {
  __shared__ float sW1[EDIM][EMB];
  __shared__ float sb1[EMB];
  __shared__ float sW2[EMB][NHEADS];
  __shared__ float sb2[NHEADS];
  const int tid = threadIdx.x;
  for (int i = tid; i < EDIM * EMB; i += 256) sW1[i / EMB][i % EMB] = We1[i];
  for (int i = tid; i < EMB; i += 256) sb1[i] = be1[i];
  for (int i = tid; i < EMB * NHEADS; i += 256) sW2[i / NHEADS][i % NHEADS] = We2[i];
  if (tid < NHEADS) sb2[tid] = be2[tid];
  __syncthreads();

  const int wave = tid >> 5, lane = tid & 31;
  const int sub  = lane >> 3;   // edge within wave
  const int h    = lane & 7;    // head for this lane
  const int e    = blockIdx.x * 32 + wave * 4 + sub;
  const int src  = (int)ei[e];
  const int tgt  = (int)ei[NEDGES + e];

  float ea[EDIM];
#pragma unroll
  for (int k = 0; k < EDIM; ++k) ea[k] = edge_attr[(size_t)e * EDIM + k];

  // hidden units j in [16h, 16h+16); partial second-layer sums for all heads
  float bp[NHEADS];
#pragma unroll
  for (int hh = 0; hh < NHEADS; ++hh) bp[hh] = 0.0f;
  const int jb = h * 16;
#pragma unroll
  for (int jj = 0; jj < 16; ++jj) {
    const int j = jb + jj;
    float z = sb1[j];
#pragma unroll
    for (int k = 0; k < EDIM; ++k) z += ea[k] * sW1[k][j];
    const float g = 0.5f * z * (1.0f + erff(z * 0.70710678118654752f));  // exact GELU
#pragma unroll
    for (int hh = 0; hh < NHEADS; ++hh) bp[hh] += g * sW2[j][hh];
  }
  // reduce across the 8 lanes of this edge (xor masks < 8 stay in-group)
#pragma unroll
  for (int m = 1; m < 8; m <<= 1)
#pragma unroll
    for (int hh = 0; hh < NHEADS; ++hh) bp[hh] += __shfl_xor(bp[hh], m, 32);
  const float bias = bp[h] + sb2[h];

  const float* qp = Q + (size_t)tgt * EMB + h * HDIM;
  const float* kp = K + (size_t)src * EMB + h * HDIM;
  float dot = 0.0f;
#pragma unroll
  for (int d = 0; d < HDIM; ++d) dot += qp[d] * kp[d];

  const float score = dot * 0.25f + bias;  // 1/sqrt(16)
  attn[(size_t)e * NHEADS + h] = score;
  atomicMax(&segmax[(size_t)tgt * NHEADS + h], fmap(score));
}

// ---------------------------------------------------------------------------
// Pass 2: ex = exp(score - segmax[tgt]) in place; denom += ex (atomic).
// ---------------------------------------------------------------------------
__global__ __launch_bounds__(256) void softmax_exp_kernel(
    const long long* __restrict__ ei, float* __restrict__ attn,
    const unsigned* __restrict__ segmax, float* __restrict__ denom) {
  const int idx = blockIdx.x * 256 + threadIdx.x;  // < E*8
  const int e = idx >> 3, h = idx & 7;
  const int tgt = (int)ei[NEDGES + e];
  const float m  = funmap(segmax[(size_t)tgt * NHEADS + h]);
  const float ex = __expf(attn[idx] - m);
  attn[idx] = ex;
  atomicAdd(&denom[(size_t)tgt * NHEADS + h], ex);
}

// ---------------------------------------------------------------------------
// Pass 3: one wave per edge. Lanes 0-7 normalize (w = ex/denom) and commit
// attn_weights output in place; all lanes scatter w * V[src] (float4 per
// lane, 4 dims -> head = lane>>2 broadcast via shfl) into accum via
// GLOBAL_ATOMIC_ADD_F32.
// ---------------------------------------------------------------------------
__global__ __launch_bounds__(256) void weighted_scatter_kernel(
    const long long* __restrict__ ei, float* __restrict__ attn,
    const float* __restrict__ denom, const float* __restrict__ V,
    float* __restrict__ acc) {
  const int tid  = threadIdx.x;
  const int e    = blockIdx.x * 8 + (tid >> 5);
  const int lane = tid & 31;
  const int src  = (int)ei[e];
  const int tgt  = (int)ei[NEDGES + e];

  float w = 0.0f;
  if (lane < NHEADS) {
    const float ex  = attn[(size_t)e * NHEADS + lane];
    const float den = denom[(size_t)tgt * NHEADS + lane];
    w = ex / den;
    attn[(size_t)e * NHEADS + lane] = w;  // final attn_weights output
  }
  const float wv = __shfl(w, lane >> 2, 32);  // head of this lane's 4 dims

  const float4 v = *(const float4*)(V + (size_t)src * EMB + lane * 4);
  float* ap = acc + (size_t)tgt * EMB + lane * 4;
  atomicAdd(ap + 0, wv * v.x);
  atomicAdd(ap + 1, wv * v.y);
  atomicAdd(ap + 2, wv * v.z);
  atomicAdd(ap + 3, wv * v.w);
}

// ---------------------------------------------------------------------------
// Launch. Inputs: x, edge_index, edge_attr, Wq, Wk, Wv, We1, be1, We2, be2,
// Wo, bo. Output: [out (N*128) | attn_weights (E*8)] fp32.
// Workspace layout (floats): Q | K | V | ACC | segmax(u32) | denom  (~212 MB)
// ---------------------------------------------------------------------------
extern "C" void kernel_launch(void* const* d_in, const int* in_sizes, int n_in,
                              void* d_out, int out_size, void* d_ws, size_t ws_size,
                              hipStream_t stream) {
  const float*     x         = (const float*)d_in[0];
  const long long* ei        = (const long long*)d_in[1];  // int64 edge_index
  const float*     edge_attr = (const float*)d_in[2];
  const float*     Wq        = (const float*)d_in[3];
  const float*     Wk        = (const float*)d_in[4];
  const float*     Wv        = (const float*)d_in[5];
  const float*     We1       = (const float*)d_in[6];
  const float*     be1       = (const float*)d_in[7];
  const float*     We2       = (const float*)d_in[8];
  const float*     be2       = (const float*)d_in[9];
  const float*     Wo        = (const float*)d_in[10];
  const float*     bo        = (const float*)d_in[11];

  float* out  = (float*)d_out;                    // N*128
  float* attn = out + (size_t)NNODES * EMB;       // E*8

  const size_t NF  = (size_t)NNODES * EMB;        // 12.8M floats
  const size_t NHN = (size_t)NNODES * NHEADS;     // 0.8M
  float*    Q      = (float*)d_ws;
  float*    K      = Q + NF;
  float*    V      = K + NF;
  float*    ACC    = V + NF;
  unsigned* segmax = (unsigned*)(ACC + NF);
  float*    denom  = (float*)(segmax + NHN);

  // zero ACC | segmax | denom (contiguous) every call — graph-capturable
  hipMemsetAsync(ACC, 0, (NF + 2 * NHN) * sizeof(float), stream);

  const int gemmBlocks = NNODES / 16;  // 6250 (N divisible by 16)
  gemm128_wmma<<<gemmBlocks, 256, 0, stream>>>(x, Wq, nullptr, Q);
  gemm128_wmma<<<gemmBlocks, 256, 0, stream>>>(x, Wk, nullptr, K);
  gemm128_wmma<<<gemmBlocks, 256, 0, stream>>>(x, Wv, nullptr, V);

  edge_score_kernel<<<NEDGES / 32, 256, 0, stream>>>(
      ei, edge_attr, Q, K, We1, be1, We2, be2, attn, segmax);

  softmax_exp_kernel<<<NEDGES * NHEADS / 256, 256, 0, stream>>>(
      ei, attn, segmax, denom);

  weighted_scatter_kernel<<<NEDGES / 8, 256, 0, stream>>>(
      ei, attn, denom, V, ACC);

  gemm128_wmma<<<gemmBlocks, 256, 0, stream>>>(ACC, Wo, bo, out);
}